// LoRA_927712936104
// MI455X (gfx1250) — compile-verified
//
#include <hip/hip_runtime.h>

// ---------------------------------------------------------------------------
// Multi-adapter LoRA on gfx1250 (MI455X), f32 WMMA path.
//   out = base + 2.0 * (x @ A_a^T) @ B_a   with a = adapter_ids[t]
// Tokens are grouped by adapter so each token is computed against exactly one
// adapter (reference computes all 8 and masks; same math, 1/8 the FLOPs).
// ---------------------------------------------------------------------------

#define N_ADAPTERS 8
#define RANK       64
#define T_TOKENS   8192
#define DIN        4096
#define DOUT       4096
#define TILES      (T_TOKENS / 16)          // 512 16-token tiles per adapter slot
#define SCALING    2.0f

typedef __attribute__((ext_vector_type(2))) float v2f;
typedef __attribute__((ext_vector_type(8))) float v8f;

// ---------------------------------------------------------------- grouping --
__global__ void lora_zero_counts(int* counts) {
    if (threadIdx.x < N_ADAPTERS) counts[threadIdx.x] = 0;
}

__global__ void lora_scatter(const int* __restrict__ ids,
                             int* __restrict__ counts,
                             int* __restrict__ idxlist) {
    int t = blockIdx.x * blockDim.x + threadIdx.x;
    if (t < T_TOKENS) {
        int a = ids[t] & (N_ADAPTERS - 1);
        int p = atomicAdd(&counts[a], 1);
        idxlist[a * T_TOKENS + p] = t;
    }
}

// ---------------------------------------------------------------- stage 1 ---
// xa[t, 0:64] = SCALING * x[t, :] @ A_a[:, :]^T for the 16 tokens of one tile.
// Block = 128 threads = 4 waves; wave w computes rank columns [16w, 16w+16).
// K loop: 4096 in steps of 4 -> 1024 x v_wmma_f32_16x16x4_f32 per wave.
__global__ void __launch_bounds__(128)
lora_stage1(const float* __restrict__ x,
            const float* __restrict__ loraA,
            const int*   __restrict__ counts,
            const int*   __restrict__ idxlist,
            float*       __restrict__ xa) {
    const int slot = blockIdx.x;
    const int a    = slot / TILES;
    const int tile = slot % TILES;
    const int cnt  = counts[a];
    const int base = tile * 16;
    if (base >= cnt) return;                       // uniform per block

    __shared__ int tok[16];
    if (threadIdx.x < 16) {
        int p = base + threadIdx.x;
        tok[threadIdx.x] = (p < cnt) ? idxlist[a * T_TOKENS + p]
                                     : idxlist[a * T_TOKENS + base]; // dup pad
    }
    __syncthreads();

    const int lane      = threadIdx.x & 31;
    const int wave      = threadIdx.x >> 5;
    const int rank_base = wave * 16;

    // A-operand (16x4 f32): lanes 0-15 hold K={0,1}, lanes 16-31 hold K={2,3}
    const int row  = lane & 15;
    const int koff = (lane >> 4) * 2;
    const float* xrow = x + (size_t)tok[row] * DIN;
    // B-operand (4x16 f32): same K split; N = lane&15 selects rank column
    const int rank = rank_base + (lane & 15);
    const float* arow = loraA + ((size_t)(a * RANK + rank)) * DIN;

    v8f c = {};
#pragma unroll 4
    for (int k = 0; k < DIN; k += 4) {
        v2f av, bv;
        av.x = xrow[k + koff];
        av.y = xrow[k + koff + 1];
        bv.x = arow[k + koff];
        bv.y = arow[k + koff + 1];
        c = __builtin_amdgcn_wmma_f32_16x16x4_f32(
                false, av, false, bv, (short)0, c, false, false);
    }

    // D layout: VGPR v -> M = v + 8*(lane>>4), N = lane&15
    const int n    = lane & 15;
    const int mofs = (lane >> 4) * 8;
#pragma unroll
    for (int v = 0; v < 8; ++v) {
        int m = v + mofs;
        if (base + m < cnt)
            xa[(size_t)tok[m] * RANK + rank_base + n] = c[v] * SCALING;
    }
}

// ---------------------------------------------------------------- stage 2 ---
// out[t, col] = base_output[t, col] + xa[t, 0:64] @ B_a[0:64, col]
// Block = 256 threads = 8 waves; wave w covers 16 output columns.
// grid.y = DOUT / 128 column chunks. K loop: 64 in steps of 4 -> 16 WMMAs.
__global__ void __launch_bounds__(256)
lora_stage2(const float* __restrict__ xa,
            const float* __restrict__ loraB,
            const float* __restrict__ base_out,
            const int*   __restrict__ counts,
            const int*   __restrict__ idxlist,
            float*       __restrict__ out) {
    const int slot = blockIdx.x;
    const int a    = slot / TILES;
    const int tile = slot % TILES;
    const int cnt  = counts[a];
    const int base = tile * 16;
    if (base >= cnt) return;

    __shared__ int tok[16];
    if (threadIdx.x < 16) {
        int p = base + threadIdx.x;
        tok[threadIdx.x] = (p < cnt) ? idxlist[a * T_TOKENS + p]
                                     : idxlist[a * T_TOKENS + base];
    }
    __syncthreads();

    const int lane = threadIdx.x & 31;
    const int wave = threadIdx.x >> 5;
    const int col  = blockIdx.y * 128 + wave * 16 + (lane & 15);

    const int row  = lane & 15;
    const int koff = (lane >> 4) * 2;
    const float* xar = xa + (size_t)tok[row] * RANK;

    v8f c = {};
#pragma unroll
    for (int k = 0; k < RANK; k += 4) {
        v2f av, bv;
        av.x = xar[k + koff];
        av.y = xar[k + koff + 1];
        const float* bp = loraB + ((size_t)(a * RANK + k + koff)) * DOUT + col;
        bv.x = bp[0];
        bv.y = bp[DOUT];
        c = __builtin_amdgcn_wmma_f32_16x16x4_f32(
                false, av, false, bv, (short)0, c, false, false);
    }

    const int mofs = (lane >> 4) * 8;
#pragma unroll
    for (int v = 0; v < 8; ++v) {
        int m = v + mofs;
        if (base + m < cnt) {
            size_t o = (size_t)tok[m] * DOUT + col;
            out[o] = base_out[o] + c[v];
        }
    }
}

// ---------------------------------------------------------------- launcher --
extern "C" void kernel_launch(void* const* d_in, const int* in_sizes, int n_in,
                              void* d_out, int out_size, void* d_ws, size_t ws_size,
                              hipStream_t stream) {
    const float* x     = (const float*)d_in[0];
    const float* loraA = (const float*)d_in[1];
    const float* loraB = (const float*)d_in[2];
    const float* baseo = (const float*)d_in[3];
    const int*   ids   = (const int*)d_in[4];
    float*       out   = (float*)d_out;

    // Workspace layout: [counts: 8 ints][pad to 256B][idxlist: 8*8192 ints][xa: 8192*64 f32]
    int*   counts  = (int*)d_ws;
    int*   idxlist = (int*)((char*)d_ws + 256);
    float* xa      = (float*)((char*)d_ws + 256 + (size_t)N_ADAPTERS * T_TOKENS * sizeof(int));

    lora_zero_counts<<<1, 32, 0, stream>>>(counts);
    lora_scatter<<<T_TOKENS / 256, 256, 0, stream>>>(ids, counts, idxlist);
    lora_stage1<<<dim3(N_ADAPTERS * TILES), 128, 0, stream>>>(x, loraA, counts, idxlist, xa);
    lora_stage2<<<dim3(N_ADAPTERS * TILES, DOUT / 128), 256, 0, stream>>>(
        xa, loraB, baseo, counts, idxlist, out);
}